// GRAPH_AUTOENCODER_32899449488048
// MI455X (gfx1250) — compile-verified
//
#include <hip/hip_runtime.h>
#include <math.h>

// ---------------- problem constants (match reference) ----------------
#define NN        50000      // nodes
#define NG        500        // graphs
#define NPGC      100        // nodes per graph
#define E_MAIN    800000
#define EPG_MAIN  1600
#define E_SUB     400000
#define EPG_SUB   800
#define F0C       128
#define H1C       256
#define H2C       128
#define BNEPS     1e-5f

typedef float v2f __attribute__((ext_vector_type(2)));
typedef float v8f __attribute__((ext_vector_type(8)));

// =====================================================================
// CSR build: one block per graph. Edges of graph g are [g*epg,(g+1)*epg),
// dst nodes are in [g*100, g*100+100). Integer LDS atomics for counts
// (deterministic), fixed-order bucket placement (deterministic fp sums
// later). Also produces dinv[n] = rsqrt(indeg+1).
// =====================================================================
__global__ void build_csr(const int* __restrict__ src, const int* __restrict__ dst,
                          int epg, int* __restrict__ ptr, int* __restrict__ slist,
                          float* __restrict__ dinv, int totalE) {
    __shared__ int cnt[NPGC];
    __shared__ int offs[NPGC];
    extern __shared__ int dl[];                  // epg ints: local dst per edge
    const int g = blockIdx.x, t = threadIdx.x;
    const int ebase = g * epg, nbase = g * NPGC;
    for (int i = t; i < NPGC; i += blockDim.x) cnt[i] = 0;
    __syncthreads();
    for (int e = t; e < epg; e += blockDim.x) {
        int d = dst[ebase + e] - nbase;
        dl[e] = d;
        atomicAdd(&cnt[d], 1);
    }
    __syncthreads();
    if (t == 0) {
        int run = ebase;
        for (int i = 0; i < NPGC; i++) { ptr[nbase + i] = run; offs[i] = run; run += cnt[i]; }
        if (g == NG - 1) ptr[NN] = totalE;
    }
    __syncthreads();
    for (int i = t; i < NPGC; i += blockDim.x)
        dinv[nbase + i] = rsqrtf((float)cnt[i] + 1.0f);
    // fixed-order placement: thread per destination bucket scans edges in order
    for (int d = t; d < NPGC; d += blockDim.x) {
        int p = offs[d];
        for (int e = 0; e < epg; e++)
            if (dl[e] == d) slist[p++] = src[ebase + e];
    }
}

// =====================================================================
// fp32 WMMA GEMM:  C[M x NC] = act( A[M x K] @ W[K x NC] + bias )
// Block: 256 threads (8 wave32), 64 rows x NC cols per block.
// W staged in LDS as k-pairs: Ws2[(k/2)*LDP + c] = {W[k][c], W[k+1][c]},
// so each B fragment is ONE aligned ds_load_b64 (no register shuffles).
// LDP = NC+16 float2 elems -> hi/lo lane groups hit disjoint bank halves.
// A rows are CLAMPED (not predicated): invalid A rows only feed D rows
// >= M which are never stored -> unconditional global_load_b64, no EXEC
// manipulation in the k-loop.
// Uses V_WMMA_F32_16X16X4_F32 (full fp32 precision).
// ACT: 0=none, 1=relu, 2=sigmoid
// =====================================================================
template <int K, int NC, int ACT, int BIAS>
__global__ void gemm_wmma(const float* __restrict__ A, const float* __restrict__ W,
                          const float* __restrict__ bias, float* __restrict__ C, int M) {
    constexpr int LDP = NC + 16;                 // stride in float2 elements
    constexpr int NCT = NC / 32;                 // col tiles per wave
    extern __shared__ float lds_f[];
    v2f* Ws2 = (v2f*)lds_f;
    const int tid = threadIdx.x;
    for (int i = tid; i < (K / 2) * NC; i += 256) {
        const int k2 = i / NC, c = i - k2 * NC;
        v2f p;
        p.x = W[(2 * k2) * NC + c];
        p.y = W[(2 * k2 + 1) * NC + c];
        Ws2[k2 * LDP + c] = p;
    }
    __syncthreads();

    const int wave = tid >> 5, lane = tid & 31;
    const int rt = wave >> 1, ch = wave & 1;     // 4 row-tiles x 2 col-halves
    const int row0 = blockIdx.x * 64 + rt * 16;
    const int l16 = lane & 15;
    const bool hi = lane >= 16;

    v8f acc[NCT] = {};
    const int ar = row0 + l16;                   // A row owned by this lane
    const int arc = (ar < M) ? ar : (M - 1);     // clamp (see header comment)
    const float* Arow = A + (long)arc * K;
    const int colb = ch * NCT * 16 + l16;

    for (int kk = 0; kk < K; kk += 4) {
        // A frag (ISA 16x4 f32 layout): v0 = K=kk (lanes0-15) / kk+2 (16-31), v1 = kk+1 / kk+3
        const int ks = kk + (hi ? 2 : 0);
        const v2f a = *(const v2f*)(Arow + ks);
        const v2f* brow = Ws2 + (ks >> 1) * LDP + colb;
#pragma unroll
        for (int j = 0; j < NCT; j++) {
            const v2f b = brow[j * 16];          // rows {ks, ks+1}, col
            acc[j] = __builtin_amdgcn_wmma_f32_16x16x4_f32(
                false, a, false, b, (short)0, acc[j], false, false);
        }
    }
    // store: C/D layout: VGPR i -> row i (lanes0-15) / row i+8 (lanes16-31)
#pragma unroll
    for (int j = 0; j < NCT; j++) {
        const int col = colb + j * 16;
        const float bv = BIAS ? bias[col] : 0.0f;
#pragma unroll
        for (int i = 0; i < 8; i++) {
            const int row = row0 + i + (hi ? 8 : 0);
            if (row < M) {
                float v = acc[j][i] + bv;
                if (ACT == 1) v = fmaxf(v, 0.0f);
                else if (ACT == 2) v = 1.0f / (1.0f + __expf(-v));
                C[(long)row * NC + col] = v;
            }
        }
    }
}

// =====================================================================
// GCN combine: out[n] = sum_{e in in(n)} A[src_e]*dinv[src_e]*dinv[n]
//                        + A[n]*dinv[n]^2 + bias   (optional relu)
// grid = NN blocks, block = F threads. Deterministic (CSR order fixed).
// =====================================================================
template <int F, int ACT>
__global__ void gcn_combine(const float* __restrict__ Ain, const int* __restrict__ ptr,
                            const int* __restrict__ slist, const float* __restrict__ dinv,
                            const float* __restrict__ bias, float* __restrict__ out) {
    const int n = blockIdx.x, f = threadIdx.x;
    const int e0 = ptr[n], e1 = ptr[n + 1];
    const float di = dinv[n];
    float s = 0.0f;
    for (int e = e0; e < e1; e++) {
        const int sn = slist[e];
        s += Ain[(long)sn * F + f] * dinv[sn];
    }
    float v = s * di + Ain[(long)n * F + f] * di * di + bias[f];
    if (ACT == 1) v = fmaxf(v, 0.0f);
    out[(long)n * F + f] = v;
}

// ---------------- BatchNorm over rows (column stats), deterministic --
template <int C>
__global__ void bn_partial(const float* __restrict__ X, int Nrows, int nb,
                           float* __restrict__ psum, float* __restrict__ psq) {
    const int b = blockIdx.x, c = threadIdx.x;
    const int chunk = (Nrows + nb - 1) / nb;
    const int r0 = b * chunk;
    const int r1 = (r0 + chunk < Nrows) ? r0 + chunk : Nrows;
    float s = 0.f, q = 0.f;
    for (int r = r0; r < r1; r++) { float v = X[(long)r * C + c]; s += v; q += v * v; }
    psum[b * C + c] = s;
    psq[b * C + c] = q;
}
template <int C>
__global__ void bn_finalize(const float* __restrict__ psum, const float* __restrict__ psq,
                            int nb, int Nrows, float* __restrict__ mean, float* __restrict__ rstd) {
    const int c = threadIdx.x;
    float s = 0.f, q = 0.f;
    for (int b = 0; b < nb; b++) { s += psum[b * C + c]; q += psq[b * C + c]; }
    const float m = s / (float)Nrows;
    const float v = q / (float)Nrows - m * m;
    mean[c] = m;
    rstd[c] = rsqrtf(v + BNEPS);
}
template <int C>
__global__ void bn_apply(float* __restrict__ X, const float* __restrict__ mean,
                         const float* __restrict__ rstd, int total) {
    const int i = blockIdx.x * blockDim.x + threadIdx.x;
    if (i < total) {
        const int c = i & (C - 1);
        X[i] = (X[i] - mean[c]) * rstd[c];
    }
}

// ---------------- per-graph BN (100 contiguous rows per graph) -------
template <int C>
__global__ void seg_bn_kernel(float* __restrict__ X) {
    const int g = blockIdx.x, c = threadIdx.x;
    const long base = (long)g * NPGC * C + c;
    float s = 0.f, q = 0.f;
    for (int r = 0; r < NPGC; r++) { float v = X[base + (long)r * C]; s += v; q += v * v; }
    const float m = s * (1.0f / NPGC);
    const float var = q * (1.0f / NPGC) - m * m;
    const float rs = rsqrtf(var + BNEPS);
    for (int r = 0; r < NPGC; r++) { const long i = base + (long)r * C; X[i] = (X[i] - m) * rs; }
}

// ---------------- row-wise L2 normalize (C = 128), wave per row ------
__global__ void l2n_kernel(const float* __restrict__ X, float* __restrict__ Y, int M) {
    const int row = (blockIdx.x * blockDim.x + threadIdx.x) >> 5;
    const int lane = threadIdx.x & 31;
    if (row >= M) return;
    const float* xr = X + (long)row * H2C;
    float v0 = xr[lane], v1 = xr[lane + 32], v2 = xr[lane + 64], v3 = xr[lane + 96];
    float s = v0 * v0 + v1 * v1 + v2 * v2 + v3 * v3;
    for (int off = 16; off; off >>= 1) s += __shfl_xor(s, off, 32);
    const float scale = 1.0f / fmaxf(sqrtf(s), 1e-12f);
    float* yr = Y + (long)row * H2C;
    yr[lane] = v0 * scale; yr[lane + 32] = v1 * scale;
    yr[lane + 64] = v2 * scale; yr[lane + 96] = v3 * scale;
}

// ---------------- pooling: segment max of z and z+noise --------------
__global__ void pool_max(const float* __restrict__ Z, const float* __restrict__ noise,
                         float* __restrict__ zg_out, float* __restrict__ poolbuf) {
    const int g = blockIdx.x, f = threadIdx.x;  // 128 threads
    float m1 = -INFINITY, m2 = -INFINITY;
    for (int r = 0; r < NPGC; r++) {
        const long i = ((long)g * NPGC + r) * H2C + f;
        const float v = Z[i];
        m1 = fmaxf(m1, v);
        m2 = fmaxf(m2, v + noise[i]);
    }
    zg_out[g * H2C + f] = m1;
    poolbuf[g * H2C + f] = m1;
    poolbuf[(NG + g) * H2C + f] = m2;
}

__global__ void pool_mean(const float* __restrict__ Z, float* __restrict__ out) {
    const int g = blockIdx.x, f = threadIdx.x;
    float s = 0.f;
    for (int r = 0; r < NPGC; r++) s += Z[((long)g * NPGC + r) * H2C + f];
    out[g * H2C + f] = s * (1.0f / NPGC);
}

// =====================================================================
extern "C" void kernel_launch(void* const* d_in, const int* in_sizes, int n_in,
                              void* d_out_v, int out_size, void* d_ws, size_t ws_size,
                              hipStream_t stream) {
    const float* x        = (const float*)d_in[0];
    const int*   srcE     = (const int*)d_in[1];
    const int*   dstE     = (const int*)d_in[2];
    const float* pos_x    = (const float*)d_in[4];
    const int*   pos_src  = (const int*)d_in[5];
    const int*   pos_dst  = (const int*)d_in[6];
    const float* neg_x    = (const float*)d_in[8];
    const int*   neg_src  = (const int*)d_in[9];
    const int*   neg_dst  = (const int*)d_in[10];
    const float* target_x = (const float*)d_in[12];
    const float* noise    = (const float*)d_in[13];
    const float* W_enc1   = (const float*)d_in[14];
    const float* b_enc1   = (const float*)d_in[15];
    const float* W_enc2   = (const float*)d_in[16];
    const float* b_enc2   = (const float*)d_in[17];
    const float* W_dec1   = (const float*)d_in[18];
    const float* W_dec2   = (const float*)d_in[19];
    const float* Wn1      = (const float*)d_in[20];
    const float* bn1      = (const float*)d_in[21];
    const float* Wn2      = (const float*)d_in[22];
    const float* bn2      = (const float*)d_in[23];
    const float* Ws1      = (const float*)d_in[24];
    const float* bs1      = (const float*)d_in[25];
    const float* Wp1      = (const float*)d_in[26];
    const float* bp1      = (const float*)d_in[27];
    const float* Wp2      = (const float*)d_in[28];
    const float* bp2      = (const float*)d_in[29];
    float* out = (float*)d_out_v;

    // ---- workspace carve (floats / ints) ----
    float* bufA    = (float*)d_ws;                       // NN*H1
    float* bufB    = bufA + (long)NN * H1C;              // NN*H1
    float* bufA2   = bufA + (long)NN * H2C;              // second half of bufA (NN*H2)
    int*   csr_ptr = (int*)(bufB + (long)NN * H1C);      // NN+1
    int*   csr_src = csr_ptr + (NN + 2);                 // E_MAIN
    float* dinv    = (float*)(csr_src + E_MAIN);         // NN
    float* poolbuf = dinv + NN;                          // 2*NG*H2
    float* psum    = poolbuf + 2 * NG * H2C;             // 256*H1
    float* psq     = psum + 256 * H1C;                   // 256*H1
    float* meanv   = psq + 256 * H1C;                    // H1
    float* rstdv   = meanv + H1C;                        // H1
    float* t1      = rstdv + H1C;                        // NG*H1
    float* projtmp = t1 + (long)NG * H1C;                // 2*NG*H2

    // ---- output offsets (reference tuple order) ----
    float* o_z   = out;
    float* o_zg  = o_z + (long)NN * H2C;
    float* o_xr  = o_zg + (long)NG * H2C;
    float* o_pos = o_xr + (long)NN * F0C;
    float* o_neg = o_pos + (long)NG * H2C;
    float* o_mlp = o_neg + (long)NG * H2C;               // z_g_mlp then z'_g_mlp (contiguous)
    float* o_tz  = o_mlp + (long)2 * NG * H2C;

    constexpr int LDS_128_256 = (128 / 2) * (256 + 16) * 8;
    constexpr int LDS_256_128 = (256 / 2) * (128 + 16) * 8;
    constexpr int LDS_128_128 = (128 / 2) * (128 + 16) * 8;
    const int gbN  = (NN + 63) / 64;
    const int gb1k = (2 * NG + 63) / 64;
    const int gb5h = (NG + 63) / 64;

    // ================= main encoder =================
    build_csr<<<NG, 128, EPG_MAIN * 4, stream>>>(srcE, dstE, EPG_MAIN, csr_ptr, csr_src, dinv, E_MAIN);
    gemm_wmma<128, 256, 0, 0><<<gbN, 256, LDS_128_256, stream>>>(x, W_enc1, nullptr, bufA, NN);
    gcn_combine<256, 1><<<NN, 256, 0, stream>>>(bufA, csr_ptr, csr_src, dinv, b_enc1, bufB);
    bn_partial<256><<<256, 256, 0, stream>>>(bufB, NN, 256, psum, psq);
    bn_finalize<256><<<1, 256, 0, stream>>>(psum, psq, 256, NN, meanv, rstdv);
    bn_apply<256><<<(NN * 256 + 255) / 256, 256, 0, stream>>>(bufB, meanv, rstdv, NN * 256);
    gemm_wmma<256, 128, 0, 0><<<gbN, 256, LDS_256_128, stream>>>(bufB, W_enc2, nullptr, bufA, NN);
    gcn_combine<128, 0><<<NN, 128, 0, stream>>>(bufA, csr_ptr, csr_src, dinv, b_enc2, bufA2);
    l2n_kernel<<<(NN + 7) / 8, 256, 0, stream>>>(bufA2, o_z, NN);           // z
    pool_max<<<NG, 128, 0, stream>>>(o_z, noise, o_zg, poolbuf);            // z_g, z'_g

    // ================= decoder =================
    gemm_wmma<128, 256, 1, 0><<<gbN, 256, LDS_128_256, stream>>>(o_z, W_dec1, nullptr, bufB, NN);
    gemm_wmma<256, 128, 2, 0><<<gbN, 256, LDS_256_128, stream>>>(bufB, W_dec2, nullptr, o_xr, NN);

    // ================= projection head (z_g and z'_g batched, M=1000) =====
    gemm_wmma<128, 128, 1, 1><<<gb1k, 256, LDS_128_128, stream>>>(poolbuf, Wp1, bp1, projtmp, 2 * NG);
    gemm_wmma<128, 128, 0, 1><<<gb1k, 256, LDS_128_128, stream>>>(projtmp, Wp2, bp2, o_mlp, 2 * NG);

    // ================= pos / neg subgraph branches =================
    for (int branch = 0; branch < 2; branch++) {
        const float* sx = branch ? neg_x : pos_x;
        const int* ss   = branch ? neg_src : pos_src;
        const int* sd   = branch ? neg_dst : pos_dst;
        float* o_sub    = branch ? o_neg : o_pos;
        build_csr<<<NG, 128, EPG_SUB * 4, stream>>>(ss, sd, EPG_SUB, csr_ptr, csr_src, dinv, E_SUB);
        gemm_wmma<128, 256, 0, 0><<<gbN, 256, LDS_128_256, stream>>>(sx, Ws1, nullptr, bufA, NN);
        gcn_combine<256, 1><<<NN, 256, 0, stream>>>(bufA, csr_ptr, csr_src, dinv, bs1, bufB);
        seg_bn_kernel<256><<<NG, 256, 0, stream>>>(bufB);
        gemm_wmma<256, 128, 0, 0><<<gbN, 256, LDS_256_128, stream>>>(bufB, W_enc2, nullptr, bufA, NN);
        gcn_combine<128, 0><<<NN, 128, 0, stream>>>(bufA, csr_ptr, csr_src, dinv, b_enc2, bufA2);
        l2n_kernel<<<(NN + 7) / 8, 256, 0, stream>>>(bufA2, bufB, NN);
        pool_mean<<<NG, 128, 0, stream>>>(bufB, o_sub);
    }

    // ================= target node encoder =================
    gemm_wmma<128, 256, 1, 1><<<gb5h, 256, LDS_128_256, stream>>>(target_x, Wn1, bn1, t1, NG);
    bn_partial<256><<<4, 256, 0, stream>>>(t1, NG, 4, psum, psq);
    bn_finalize<256><<<1, 256, 0, stream>>>(psum, psq, 4, NG, meanv, rstdv);
    bn_apply<256><<<(NG * 256 + 255) / 256, 256, 0, stream>>>(t1, meanv, rstdv, NG * 256);
    gemm_wmma<256, 128, 0, 1><<<gb5h, 256, LDS_256_128, stream>>>(t1, Wn2, bn2, projtmp, NG);
    l2n_kernel<<<(NG + 7) / 8, 256, 0, stream>>>(projtmp, o_tz, NG);
}